// BitpackedBernoulli_63050119905850
// MI455X (gfx1250) — compile-verified
//
#include <hip/hip_runtime.h>
#include <stdint.h>

// BitpackedBernoulli: replicate jax.random.uniform(threefry2x32, seed=1234)
// over a flat [N*count*8] array, compare against per-row probs, pack 8 bits
// -> byte MSB-first. JAX splits the iota counter array in half, so one
// threefry block (x0=i, x1=i+H) produces output elements i (v0) and i+H (v1):
// every computed word is used.

static constexpr uint32_t N_PROBS    = 32768u;
static constexpr uint32_t COUNT      = 256u;
static constexpr uint32_t TOTAL_ELEM = N_PROBS * COUNT * 8u;   // 2^26 uniforms
static constexpr uint32_t HALF_ELEM  = TOTAL_ELEM / 2u;        // 2^25
static constexpr uint32_t HALF_BYTES = N_PROBS * COUNT / 2u;   // 4,194,304

// Threefry-2x32 with the key baked in at compile time (jax.random.key(1234)
// -> k0 = 0, k1 = 1234). Rotations and key-injection schedule match
// jax._src.prng.threefry2x32 exactly.
__device__ __forceinline__ void threefry2x32_seed1234(uint32_t c0, uint32_t c1,
                                                      uint32_t& o0, uint32_t& o1) {
  constexpr uint32_t KS0 = 0u;
  constexpr uint32_t KS1 = 1234u;
  constexpr uint32_t KS2 = 0x1BD11BDAu ^ KS0 ^ KS1;
  uint32_t x0 = c0 + KS0;
  uint32_t x1 = c1 + KS1;
#define TF_ROUND(r)                                   \
  {                                                   \
    x0 += x1;                                         \
    x1 = (x1 << (r)) | (x1 >> (32 - (r)));  /* v_alignbit_b32 */ \
    x1 ^= x0;                                         \
  }
  TF_ROUND(13) TF_ROUND(15) TF_ROUND(26) TF_ROUND(6)
  x0 += KS1; x1 += KS2 + 1u;
  TF_ROUND(17) TF_ROUND(29) TF_ROUND(16) TF_ROUND(24)
  x0 += KS2; x1 += KS0 + 2u;
  TF_ROUND(13) TF_ROUND(15) TF_ROUND(26) TF_ROUND(6)
  x0 += KS0; x1 += KS1 + 3u;
  TF_ROUND(17) TF_ROUND(29) TF_ROUND(16) TF_ROUND(24)
  x0 += KS1; x1 += KS2 + 4u;
  TF_ROUND(13) TF_ROUND(15) TF_ROUND(26) TF_ROUND(6)
  x0 += KS2; x1 += KS0 + 5u;
#undef TF_ROUND
  o0 = x0;
  o1 = x1;
}

// JAX uniform: u = bitcast((bits >> 9) | 0x3f800000) - 1.0f  (exact sub),
// then Bernoulli bit = (u < p). Do NOT fold the -1.0f into the compare:
// (1.0f + p) rounds and changes results near ulp boundaries.
__device__ __forceinline__ uint32_t bern_bit(uint32_t v, float p) {
  float u = __uint_as_float((v >> 9) | 0x3f800000u) - 1.0f;
  return (u < p) ? 1u : 0u;
}

__global__ __launch_bounds__(256) void
BitpackedBernoulli_63050119905850_kernel(const float* __restrict__ probs,
                                         float* __restrict__ out) {
  // Thread t owns output byte t (first half) and byte t + HALF_BYTES (second
  // half). Byte t covers flat elements [8t, 8t+8); its probability row is
  // t / COUNT == blockIdx.x (block-uniform -> scalar load). Second-half byte
  // sits at row blockIdx.x + N_PROBS/2.
  const uint32_t t = blockIdx.x * 256u + threadIdx.x;
  const float p0 = probs[blockIdx.x];                   // s_load: uniform
  const float p1 = probs[blockIdx.x + (N_PROBS / 2u)];  // s_load: uniform
  const uint32_t base = t * 8u;

  uint32_t v0[8], v1[8];
#pragma unroll
  for (int k = 0; k < 8; ++k) {
    // 8 independent ARX chains -> good ILP after unroll.
    threefry2x32_seed1234(base + (uint32_t)k, base + (uint32_t)k + HALF_ELEM,
                          v0[k], v1[k]);
  }

  // MSB-first packbits: k=0 ends up at bit 7 (weight 128).
  uint32_t b0 = 0u, b1 = 0u;
#pragma unroll
  for (int k = 0; k < 8; ++k) {
    b0 = (b0 << 1) | bern_bit(v0[k], p0);
    b1 = (b1 << 1) | bern_bit(v1[k], p1);
  }

  // Harness compares non-bf16 outputs as float32: write byte values as f32.
  out[t] = (float)b0;
  out[t + HALF_BYTES] = (float)b1;
}

extern "C" void kernel_launch(void* const* d_in, const int* in_sizes, int n_in,
                              void* d_out, int out_size, void* d_ws, size_t ws_size,
                              hipStream_t stream) {
  (void)n_in; (void)d_ws; (void)ws_size;
  const float* probs = (const float*)d_in[0];   // [1, 32768] float32
  (void)in_sizes;                               // count==256 baked in (d_in[1])
  float* out = (float*)d_out;                   // 8,388,608 packed-byte values

  // One thread per first-half byte: out_size/2 threads, 256 per block.
  const int threads = 256;
  const int blocks = (out_size / 2 + threads - 1) / threads;  // 16384
  BitpackedBernoulli_63050119905850_kernel<<<blocks, threads, 0, stream>>>(probs, out);
}